// Must3r_16183436771438
// MI455X (gfx1250) — compile-verified
//
#include <hip/hip_runtime.h>
#include <math.h>

// ---------------------------------------------------------------------------
// Model constants (from the reference)
// ---------------------------------------------------------------------------
#define Lc   12
#define Dc   768
#define Hc   12
#define HDc  64
#define DEc  1024
#define MLPc 3072
#define Bc   2
#define Gc   24
#define Nc   (Gc*Gc)      // 576
#define Sc   (Nc+1)       // 577
#define S2c  (2*Sc)       // 1154

typedef float  v2f __attribute__((ext_vector_type(2)));
typedef float  v8f __attribute__((ext_vector_type(8)));

// ---------------------------------------------------------------------------
// WMMA fp32 GEMM:  C[M,N] (+)= act( A[M,K] @ W[K,N] + bias )
// One wave computes a 32x16 C tile (two 16x16 sub-tiles stacked in M) via
// V_WMMA_F32_16X16X4_F32, reusing each B fragment for two WMMAs (halves the
// dominant weight-matrix traffic per FLOP).  Block = 4 waves -> 32x64 tile.
// A tile (32x32) staged in LDS; next W chunk prefetched (global_prefetch_b8).
// Row remap for output: orow = (m/rpbIn)*rpbOut + rowOff + (m%rpbIn)
// act: 0 = none, 1 = exact GELU.  accum: 1 -> C += result.
// N must be a multiple of 64, K a multiple of 32.
// ---------------------------------------------------------------------------
__global__ void __launch_bounds__(128)
k_gemm_wmma(const float* __restrict__ A, const float* __restrict__ W,
            const float* __restrict__ bias, float* __restrict__ C,
            int M, int Nn, int K,
            int rpbIn, int rpbOut, int rowOff,
            int act, int accum)
{
    __shared__ float As[32][33];           // 32 rows x 32 k (padded)

    const int tid  = threadIdx.x;
    const int wave = tid >> 5;
    const int lane = tid & 31;
    const int m0   = blockIdx.y * 32;
    const int n0   = blockIdx.x * 64 + wave * 16;
    const int col  = n0 + (lane & 15);
    const int kh   = (lane >> 4) * 2;      // 0 or 2 (K pair select)
    const int arow = lane & 15;

    v8f acc0 = {};
    v8f acc1 = {};

    for (int kb = 0; kb < K; kb += 32) {
        // cooperative A tile load (32 rows x 32 k), zero-padded past M
        {
            int r   = tid >> 2;            // 0..31
            int kk  = (tid & 3) * 8;       // 0,8,16,24
            int row = m0 + r;
            float4 av0, av1;
            if (row < M) {
                const float* ap = A + (size_t)row * K + kb + kk;
                av0 = *(const float4*)(ap);
                av1 = *(const float4*)(ap + 4);
            } else {
                av0 = make_float4(0.f, 0.f, 0.f, 0.f);
                av1 = av0;
            }
            As[r][kk+0] = av0.x; As[r][kk+1] = av0.y;
            As[r][kk+2] = av0.z; As[r][kk+3] = av0.w;
            As[r][kk+4] = av1.x; As[r][kk+5] = av1.y;
            As[r][kk+6] = av1.z; As[r][kk+7] = av1.w;
        }
        // speculative prefetch of next W chunk for this wave's columns
        if (kb + 32 < K) {
            __builtin_prefetch(W + (size_t)(kb + 32 + kh) * Nn + col, 0, 1);
            __builtin_prefetch(W + (size_t)(kb + 48 + kh) * Nn + col, 0, 1);
        }
        __syncthreads();

        #pragma unroll
        for (int ks = 0; ks < 32; ks += 4) {
            const int kk = ks + kh;
            v2f a0, a1, b;
            a0.x = As[arow][kk];
            a0.y = As[arow][kk + 1];
            a1.x = As[arow + 16][kk];
            a1.y = As[arow + 16][kk + 1];
            const float* wp = W + (size_t)(kb + kk) * Nn + col;
            b.x = wp[0];
            b.y = wp[Nn];
            acc0 = __builtin_amdgcn_wmma_f32_16x16x4_f32(
                       false, a0, false, b, (short)0, acc0, false, false);
            acc1 = __builtin_amdgcn_wmma_f32_16x16x4_f32(
                       false, a1, false, b, (short)0, acc1, false, false);
        }
        __syncthreads();
    }

    // store: lanes 0-15 -> rows +0..7 ; lanes 16-31 -> rows +8..15 per sub-tile
    const float bv = bias ? bias[col] : 0.0f;
    const int mB0 = m0 + (lane >> 4) * 8;
    const int mB1 = m0 + 16 + (lane >> 4) * 8;
    #pragma unroll
    for (int r = 0; r < 8; ++r) {
        int m = mB0 + r;
        if (m < M) {
            int orow = (m / rpbIn) * rpbOut + rowOff + (m % rpbIn);
            size_t idx = (size_t)orow * Nn + col;
            float val = acc0[r] + bv;
            if (act == 1)
                val = 0.5f * val * (1.0f + erff(val * 0.70710678118654752f));
            if (accum) C[idx] += val;
            else       C[idx]  = val;
        }
    }
    #pragma unroll
    for (int r = 0; r < 8; ++r) {
        int m = mB1 + r;
        if (m < M) {
            int orow = (m / rpbIn) * rpbOut + rowOff + (m % rpbIn);
            size_t idx = (size_t)orow * Nn + col;
            float val = acc1[r] + bv;
            if (act == 1)
                val = 0.5f * val * (1.0f + erff(val * 0.70710678118654752f));
            if (accum) C[idx] += val;
            else       C[idx]  = val;
        }
    }
}

// ---------------------------------------------------------------------------
// LayerNorm: one block (256 thr) per row of D elements
// ---------------------------------------------------------------------------
__global__ void __launch_bounds__(256)
k_ln(const float* __restrict__ x, const float* __restrict__ w,
     const float* __restrict__ b, float* __restrict__ out, int D)
{
    const int row = blockIdx.x;
    const float* xr = x + (size_t)row * D;
    float* orow = out + (size_t)row * D;
    __shared__ float red[256];
    const int tid = threadIdx.x;

    float s = 0.f;
    for (int i = tid; i < D; i += 256) s += xr[i];
    red[tid] = s; __syncthreads();
    for (int st = 128; st > 0; st >>= 1) { if (tid < st) red[tid] += red[tid+st]; __syncthreads(); }
    const float mean = red[0] / (float)D;
    __syncthreads();

    float vs = 0.f;
    for (int i = tid; i < D; i += 256) { float d = xr[i] - mean; vs += d*d; }
    red[tid] = vs; __syncthreads();
    for (int st = 128; st > 0; st >>= 1) { if (tid < st) red[tid] += red[tid+st]; __syncthreads(); }
    const float rstd = rsqrtf(red[0] / (float)D + 1e-6f);

    for (int i = tid; i < D; i += 256)
        orow[i] = (xr[i] - mean) * rstd * w[i] + b[i];
}

// ---------------------------------------------------------------------------
// RoPE-2D in-place on [B,S,H,64]-shaped data inside tokens of stride
// strideTok (float units) at offset `off`.  grid=(S,B), block=H*16=192.
// Head dim 64 split into two 32-halves, each a 16-pair rope with pos[...,half].
// ---------------------------------------------------------------------------
__global__ void __launch_bounds__(192)
k_rope2d(float* __restrict__ x, const int* __restrict__ pos,
         int S, int strideTok, int off)
{
    const int s  = blockIdx.x, bb = blockIdx.y;
    const int t  = threadIdx.x;
    const int h  = t >> 4, i = t & 15;
    float* base = x + (size_t)(bb * S + s) * strideTok + off + h * HDc;
    const int* pp = pos + ((size_t)bb * S + s) * 2;
    const float invf = expf(-((float)i / 16.0f) * 4.6051701859880914f); // 100^(-i/16)
    #pragma unroll
    for (int half = 0; half < 2; ++half) {
        const float f = (float)pp[half] * invf;
        const float c = cosf(f), sn = sinf(f);
        float* xb = base + half * 32;
        const float x0 = xb[i], x1 = xb[i + 16];
        xb[i]      = x0 * c - x1 * sn;
        xb[i + 16] = x1 * c + x0 * sn;
    }
}

// ---------------------------------------------------------------------------
// RoPE-1D over full D=768, per-batch integer position t[b].
// in: [B,S,D]; out row = (b*outSrow + outOff + s).  grid=(S,B), block=384.
// ---------------------------------------------------------------------------
__global__ void __launch_bounds__(384)
k_rope1d(const float* __restrict__ in, float* __restrict__ out,
         const int* __restrict__ t, int S, int outSrow, int outOff)
{
    const int s = blockIdx.x, bb = blockIdx.y;
    const int i = threadIdx.x;                 // 0..383
    const int half = Dc / 2;
    const float invf = expf(-((float)i / (float)half) * 4.6051701859880914f);
    const float f = (float)t[bb] * invf;
    const float c = cosf(f), sn = sinf(f);
    const float* xr = in + (size_t)(bb * S + s) * Dc;
    float* orow = out + ((size_t)bb * outSrow + outOff + s) * Dc;
    const float x0 = xr[i], x1 = xr[i + half];
    orow[i]        = x0 * c - x1 * sn;
    orow[i + half] = x1 * c + x0 * sn;
}

// ---------------------------------------------------------------------------
// Softmax attention: one block (128 thr) per (q-row, head, batch).
// Scores kept in LDS (Sk <= 1154).  Strides/offsets in float units per token.
// out is merged [B,Sq,H*64].
// ---------------------------------------------------------------------------
#define MAXSK 1216
__global__ void __launch_bounds__(128)
k_attn(const float* __restrict__ qb, int qStride, int qOff,
       const float* __restrict__ kb, int kStride, int kOff,
       const float* __restrict__ vb, int vStride, int vOff,
       float* __restrict__ out, int Sq, int Sk, int H)
{
    const int sq = blockIdx.x, h = blockIdx.y, bbb = blockIdx.z;
    const int tid = threadIdx.x;
    __shared__ float qs[HDc];
    __shared__ float sc[MAXSK];
    __shared__ float red[128];

    const float* qp = qb + (size_t)(bbb * Sq + sq) * qStride + qOff + h * HDc;
    if (tid < HDc) qs[tid] = qp[tid] * 0.125f;     // 1/sqrt(64)
    __syncthreads();

    float lmax = -3.4e38f;
    for (int kk = tid; kk < Sk; kk += 128) {
        const float* kp = kb + (size_t)(bbb * Sk + kk) * kStride + kOff + h * HDc;
        float d = 0.f;
        #pragma unroll 8
        for (int i = 0; i < HDc; ++i) d += qs[i] * kp[i];
        sc[kk] = d;
        lmax = fmaxf(lmax, d);
    }
    red[tid] = lmax; __syncthreads();
    for (int st = 64; st > 0; st >>= 1) { if (tid < st) red[tid] = fmaxf(red[tid], red[tid+st]); __syncthreads(); }
    const float mx = red[0];
    __syncthreads();

    float lsum = 0.f;
    for (int kk = tid; kk < Sk; kk += 128) {
        const float e = expf(sc[kk] - mx);
        sc[kk] = e; lsum += e;
    }
    red[tid] = lsum; __syncthreads();
    for (int st = 64; st > 0; st >>= 1) { if (tid < st) red[tid] += red[tid+st]; __syncthreads(); }
    const float inv = 1.0f / red[0];
    __syncthreads();

    // PV: all 128 threads active — half-block handles even kk, half odd kk
    {
        const int d  = tid & 63;
        const int hs = tid >> 6;               // 0 or 1
        float a = 0.f;
        const float* vp = vb + (size_t)bbb * Sk * vStride + vOff + h * HDc + d;
        for (int kk = hs; kk < Sk; kk += 2)
            a += sc[kk] * vp[(size_t)kk * vStride];
        red[tid] = a;
    }
    __syncthreads();
    if (tid < HDc)
        out[((size_t)(bbb * Sq + sq) * H + h) * HDc + tid] =
            (red[tid] + red[tid + 64]) * inv;
}

// ---------------------------------------------------------------------------
// Small utility kernels
// ---------------------------------------------------------------------------
__global__ void k_avg(const float* a, const float* b, float* o, int n)
{
    int i = blockIdx.x * blockDim.x + threadIdx.x;
    if (i < n) o[i] = (a[i] + b[i]) * 0.5f;
}

__global__ void k_addvec(const float* a, const float* b, float* o, int n)
{
    int i = threadIdx.x;
    if (i < n) o[i] = a[i] + b[i];
}

// write pose token into row 0 of each batch
__global__ void k_pose(const float* __restrict__ pt, float* __restrict__ c, int idx)
{
    int bb = blockIdx.x;
    for (int i = threadIdx.x; i < Dc; i += blockDim.x)
        c[(size_t)bb * Sc * Dc + i] = pt[(size_t)idx * Dc + i];
}

// p[b,0]=(0,0); p[b,s>=1] = posIn[b,s-1]+1
__global__ void k_build_pos(const int* __restrict__ posIn, int* __restrict__ p)
{
    int bb = blockIdx.x, s = threadIdx.x;
    if (s >= Sc) return;
    int v0 = 0, v1 = 0;
    if (s > 0) {
        v0 = posIn[((size_t)bb * Nc + (s - 1)) * 2 + 0] + 1;
        v1 = posIn[((size_t)bb * Nc + (s - 1)) * 2 + 1] + 1;
    }
    p[((size_t)bb * Sc + s) * 2 + 0] = v0;
    p[((size_t)bb * Sc + s) * 2 + 1] = v1;
}

__global__ void k_concat_pos(const int* __restrict__ p1, const int* __restrict__ p2,
                             int* __restrict__ pc)
{
    int bb = blockIdx.x, s = threadIdx.x;
    if (s >= Sc) return;
    pc[((size_t)bb * S2c + s) * 2 + 0]      = p1[((size_t)bb * Sc + s) * 2 + 0];
    pc[((size_t)bb * S2c + s) * 2 + 1]      = p1[((size_t)bb * Sc + s) * 2 + 1];
    pc[((size_t)bb * S2c + Sc + s) * 2 + 0] = p2[((size_t)bb * Sc + s) * 2 + 0];
    pc[((size_t)bb * S2c + Sc + s) * 2 + 1] = p2[((size_t)bb * Sc + s) * 2 + 1];
}

// ---------------------------------------------------------------------------
// Host orchestration
// ---------------------------------------------------------------------------
extern "C" void kernel_launch(void* const* d_in, const int* in_sizes, int n_in,
                              void* d_out, int out_size, void* d_ws, size_t ws_size,
                              hipStream_t stream)
{
    (void)in_sizes; (void)n_in; (void)out_size; (void)ws_size;

    const float* f1        = (const float*)d_in[0];
    const float* f2        = (const float*)d_in[1];
    const float* W_embed   = (const float*)d_in[2];
    const float* b_embed   = (const float*)d_in[3];
    const float* img2      = (const float*)d_in[4];
    const float* pose_tok  = (const float*)d_in[5];
    const float* ln_w      = (const float*)d_in[6];
    const float* ln_b      = (const float*)d_in[7];
    const float* Wqkv      = (const float*)d_in[8];
    const float* bqkv      = (const float*)d_in[9];
    const float* Wo_self   = (const float*)d_in[10];
    const float* bo_self   = (const float*)d_in[11];
    const float* Wq        = (const float*)d_in[12];
    const float* bq        = (const float*)d_in[13];
    const float* Wk        = (const float*)d_in[14];
    const float* bk        = (const float*)d_in[15];
    const float* Wv        = (const float*)d_in[16];
    const float* bv        = (const float*)d_in[17];
    const float* Wo_cross  = (const float*)d_in[18];
    const float* bo_cross  = (const float*)d_in[19];
    const float* W1        = (const float*)d_in[20];
    const float* b1        = (const float*)d_in[21];
    const float* W2        = (const float*)d_in[22];
    const float* b2        = (const float*)d_in[23];
    const float* norm_w    = (const float*)d_in[24];
    const float* norm_b    = (const float*)d_in[25];
    const int*   pos1      = (const int*)d_in[26];
    const int*   pos2      = (const int*)d_in[27];
    const int*   t1        = (const int*)d_in[28];
    const int*   t2        = (const int*)d_in[29];
    const int*   tf        = (const int*)d_in[30];

    const size_t BSD  = (size_t)Bc * Sc * Dc;      // 886272
    const size_t B2SD = (size_t)Bc * S2c * Dc;     // 1772544

    float* ws = (float*)d_ws;
    size_t off = 0;
    auto alloc = [&](size_t n) { float* p = ws + off; off += n; return p; };

    float* c1a  = alloc(BSD);
    float* c1b  = alloc(BSD);
    float* c2a  = alloc(BSD);
    float* c2b  = alloc(BSD);
    float* cfa  = alloc(BSD);
    float* cfb  = alloc(BSD);
    float* tmpH = alloc(BSD);                       // LN outputs (x-stream)
    float* yB   = alloc(B2SD);                      // LN(y) (up to 2S tokens)
    float* qkvB = alloc((size_t)Bc * Sc * 3 * Dc);
    float* qB   = alloc(BSD);
    float* kB   = alloc(B2SD);
    float* vB   = alloc(B2SD);
    float* attnB= alloc(BSD);
    float* mlpB = alloc((size_t)Bc * Sc * MLPc);    // also aliases ff at start
    float* condB= alloc(B2SD);
    float* bias2= alloc(Dc);
    int* p1b    = (int*)alloc((size_t)Bc * Sc * 2);
    int* p2b    = (int*)alloc((size_t)Bc * Sc * 2);
    int* pcb    = (int*)alloc((size_t)Bc * S2c * 2);

    float* ffB  = mlpB;                             // [B,N,DE] scratch (alias)

    auto gemm = [&](const float* A, const float* W, const float* bias, float* C,
                    int M, int Nn, int K, int rpbIn, int rpbOut, int rowOff,
                    int act, int accum) {
        dim3 g(Nn / 64, (M + 31) / 32);
        k_gemm_wmma<<<g, 128, 0, stream>>>(A, W, bias, C, M, Nn, K,
                                           rpbIn, rpbOut, rowOff, act, accum);
    };

    // ---------------- embedding ----------------
    {
        int n = Bc * Nc * DEc;
        k_avg<<<(n + 255) / 256, 256, 0, stream>>>(f1, f2, ffB, n);
        k_addvec<<<1, Dc, 0, stream>>>(b_embed, img2, bias2, Dc);
        gemm(f1,  W_embed, b_embed, c1a, Bc * Nc, Dc, DEc, Nc, Sc, 1, 0, 0);
        gemm(f2,  W_embed, bias2,   c2a, Bc * Nc, Dc, DEc, Nc, Sc, 1, 0, 0);
        gemm(ffB, W_embed, b_embed, cfa, Bc * Nc, Dc, DEc, Nc, Sc, 1, 0, 0);
        k_pose<<<Bc, 256, 0, stream>>>(pose_tok, c1a, 0);
        k_pose<<<Bc, 256, 0, stream>>>(pose_tok, c2a, 1);
        k_pose<<<Bc, 256, 0, stream>>>(pose_tok, cfa, 2);
        k_build_pos<<<Bc, 640, 0, stream>>>(pos1, p1b);
        k_build_pos<<<Bc, 640, 0, stream>>>(pos2, p2b);
        k_concat_pos<<<Bc, 640, 0, stream>>>(p1b, p2b, pcb);
    }

    // ---------------- layers ----------------
    float *c1o = c1a, *c2o = c2a, *cfo = cfa;
    float *c1n = c1b, *c2n = c2b, *cfn = cfb;
    const int Mx = Bc * Sc;

    for (int l = 0; l < Lc; ++l) {
        const float* lnw   = ln_w + (size_t)l * 4 * Dc;
        const float* lnb   = ln_b + (size_t)l * 4 * Dc;
        const float* Wqkvl = Wqkv + (size_t)l * Dc * 3 * Dc;
        const float* bqkvl = bqkv + (size_t)l * 3 * Dc;
        const float* Wol   = Wo_self  + (size_t)l * Dc * Dc;
        const float* bol   = bo_self  + (size_t)l * Dc;
        const float* Wql   = Wq + (size_t)l * Dc * Dc;
        const float* bql   = bq + (size_t)l * Dc;
        const float* Wkl   = Wk + (size_t)l * Dc * Dc;
        const float* bkl   = bk + (size_t)l * Dc;
        const float* Wvl   = Wv + (size_t)l * Dc * Dc;
        const float* bvl   = bv + (size_t)l * Dc;
        const float* Wocl  = Wo_cross + (size_t)l * Dc * Dc;
        const float* bocl  = bo_cross + (size_t)l * Dc;
        const float* W1l   = W1 + (size_t)l * Dc * MLPc;
        const float* b1l   = b1 + (size_t)l * MLPc;
        const float* W2l   = W2 + (size_t)l * MLPc * Dc;
        const float* b2l   = b2 + (size_t)l * Dc;

        auto block = [&](float* xn, const float* yb, int Sk,
                         const int* xpos, const int* ypos) {
            // ---- self attention ----
            k_ln<<<Mx, 256, 0, stream>>>(xn, lnw + 0 * Dc, lnb + 0 * Dc, tmpH, Dc);
            gemm(tmpH, Wqkvl, bqkvl, qkvB, Mx, 3 * Dc, Dc, Sc, Sc, 0, 0, 0);
            k_rope2d<<<dim3(Sc, Bc), 192, 0, stream>>>(qkvB, xpos, Sc, 3 * Dc, 0);
            k_rope2d<<<dim3(Sc, Bc), 192, 0, stream>>>(qkvB, xpos, Sc, 3 * Dc, Dc);
            k_attn<<<dim3(Sc, Hc, Bc), 128, 0, stream>>>(
                qkvB, 3 * Dc, 0, qkvB, 3 * Dc, Dc, qkvB, 3 * Dc, 2 * Dc,
                attnB, Sc, Sc, Hc);
            gemm(attnB, Wol, bol, xn, Mx, Dc, Dc, Sc, Sc, 0, 0, 1);
            // ---- cross attention ----
            k_ln<<<Mx, 256, 0, stream>>>(xn, lnw + 2 * Dc, lnb + 2 * Dc, tmpH, Dc);
            gemm(tmpH, Wql, bql, qB, Mx, Dc, Dc, Sc, Sc, 0, 0, 0);
            k_rope2d<<<dim3(Sc, Bc), 192, 0, stream>>>(qB, xpos, Sc, Dc, 0);
            const int My = Bc * Sk;
            gemm(yb, Wkl, bkl, kB, My, Dc, Dc, Sk, Sk, 0, 0, 0);
            gemm(yb, Wvl, bvl, vB, My, Dc, Dc, Sk, Sk, 0, 0, 0);
            k_rope2d<<<dim3(Sk, Bc), 192, 0, stream>>>(kB, ypos, Sk, Dc, 0);
            k_attn<<<dim3(Sc, Hc, Bc), 128, 0, stream>>>(
                qB, Dc, 0, kB, Dc, 0, vB, Dc, 0, attnB, Sc, Sk, Hc);
            gemm(attnB, Wocl, bocl, xn, Mx, Dc, Dc, Sc, Sc, 0, 0, 1);
            // ---- MLP ----
            k_ln<<<Mx, 256, 0, stream>>>(xn, lnw + 3 * Dc, lnb + 3 * Dc, tmpH, Dc);
            gemm(tmpH, W1l, b1l, mlpB, Mx, MLPc, Dc, Sc, Sc, 0, 1, 0);
            gemm(mlpB, W2l, b2l, xn, Mx, Dc, MLPc, Sc, Sc, 0, 0, 1);
        };

        // stream 1: y = LN(c2_old)
        k_ln<<<Mx, 256, 0, stream>>>(c2o, lnw + 1 * Dc, lnb + 1 * Dc, yB, Dc);
        hipMemcpyAsync(c1n, c1o, BSD * sizeof(float), hipMemcpyDeviceToDevice, stream);
        block(c1n, yB, Sc, p1b, p2b);

        // stream 2: y = LN(c1_old)
        k_ln<<<Mx, 256, 0, stream>>>(c1o, lnw + 1 * Dc, lnb + 1 * Dc, yB, Dc);
        hipMemcpyAsync(c2n, c2o, BSD * sizeof(float), hipMemcpyDeviceToDevice, stream);
        block(c2n, yB, Sc, p2b, p1b);

        // forecast: cond = [rope1d(c1,t1); rope1d(c2,t2)], y = LN(cond),
        //           x = rope1d(cf, tf)
        k_rope1d<<<dim3(Sc, Bc), 384, 0, stream>>>(c1o, condB, t1, Sc, S2c, 0);
        k_rope1d<<<dim3(Sc, Bc), 384, 0, stream>>>(c2o, condB, t2, Sc, S2c, Sc);
        k_ln<<<Bc * S2c, 256, 0, stream>>>(condB, lnw + 1 * Dc, lnb + 1 * Dc, yB, Dc);
        k_rope1d<<<dim3(Sc, Bc), 384, 0, stream>>>(cfo, cfn, tf, Sc, Sc, 0);
        block(cfn, yB, S2c, p2b, pcb);

        // swap buffers
        float* t;
        t = c1o; c1o = c1n; c1n = t;
        t = c2o; c2o = c2n; c2n = t;
        t = cfo; cfo = cfn; cfn = t;
    }

    // ---------------- final norm + stack ----------------
    float* outp = (float*)d_out;
    k_ln<<<Mx, 256, 0, stream>>>(c1o, norm_w, norm_b, outp + 0 * BSD, Dc);
    k_ln<<<Mx, 256, 0, stream>>>(c2o, norm_w, norm_b, outp + 1 * BSD, Dc);
    k_ln<<<Mx, 256, 0, stream>>>(cfo, norm_w, norm_b, outp + 2 * BSD, Dc);
}